// Decoder_12128987644664
// MI455X (gfx1250) — compile-verified
//
#include <hip/hip_runtime.h>

#define B_ROWS 32768
#define HID    512
#define DIM    256
#define MAX_N  17
#define KEY_MID 264
#define DEC_MID 384
#define SP_MID  264
#define SPW_LD  272   // padded N for sp_W1 (17 tiles of 16)

typedef __attribute__((ext_vector_type(2))) float v2f;
typedef __attribute__((ext_vector_type(8))) float v8f;

// CDNA5 fp32 WMMA: D[16x16] += A[16x4] * B[4x16], fp32 accumulate, RNE.
__device__ __forceinline__ v8f wmma4(v2f a, v2f b, v8f c) {
  return __builtin_amdgcn_wmma_f32_16x16x4_f32(false, a, false, b, (short)0, c, false, false);
}

// ---------------------------------------------------------------------------
// Phase 0: pack weights W[K][Nin] -> P[K/2][Nout][2] (zero-pad cols >= Nin).
// A WMMA B-fragment then needs ONE global_load_b64 per lane:
//   P[(kk>>1)*2*Nout + 2*col] = {W[kk][col], W[kk+1][col]}.
// ---------------------------------------------------------------------------
__global__ __launch_bounds__(256) void pack_w_kernel(
    const float* __restrict__ W, float* __restrict__ P,
    int K, int Nin, int Nout)
{
  int idx = blockIdx.x * 256 + threadIdx.x;
  if (idx >= K * Nout) return;
  int k = idx / Nout, c = idx % Nout;
  float v = (c < Nin) ? W[k * Nin + c] : 0.f;
  P[((size_t)(k >> 1) * Nout + c) * 2 + (k & 1)] = v;
}

// ---------------------------------------------------------------------------
// Dual-M WMMA K-chunk: uniform base (k0*N) + lane-invariant 32-bit offset
// (2*col + 2*N*half) + compile-time immediates (4*N*u) -> SADDR+voffset loads.
// ---------------------------------------------------------------------------
template <int N, int K>
__device__ __forceinline__ void gemm_dual(
    const float* __restrict__ Wp, int voff,
    const float* __restrict__ aRow0, const float* __restrict__ aRow1, int kstep2half,
    v8f& acc0, v8f& acc1)
{
  for (int k0 = 0; k0 < K; k0 += 16) {
    const float* bk = Wp + (size_t)k0 * N;          // uniform per wave
    v2f a0[4], a1[4], bf[4];
    #pragma unroll
    for (int u = 0; u < 4; u++) {
      int kk = k0 + 4 * u + kstep2half;             // lane-invariant + imm
      bf[u] = *(const v2f*)(bk + (voff + 4 * N * u));
      a0[u] = *(const v2f*)(aRow0 + kk);
      a1[u] = *(const v2f*)(aRow1 + kk);
    }
    #pragma unroll
    for (int u = 0; u < 4; u++) {
      acc0 = wmma4(a0[u], bf[u], acc0);
      acc1 = wmma4(a1[u], bf[u], acc1);
    }
  }
}

// ---------------------------------------------------------------------------
// Phase 1: size_pred MLP -> per-row argmax sizes n[i]
// One block = 32 rows (two 16-row M tiles sharing B fragments), 8 waves.
// ---------------------------------------------------------------------------
#define ZT_LD 516   // 516 mod 64 == 4 -> conflict-free A-fragment LDS loads

__global__ __launch_bounds__(256) void size_pred_kernel(
    const float* __restrict__ z, const float* __restrict__ W1p, const float* __restrict__ b1,
    const float* __restrict__ g, const float* __restrict__ be,
    const float* __restrict__ W2, const float* __restrict__ b2,
    int* __restrict__ n_out)
{
  __shared__ float zt[32][ZT_LD];
  __shared__ float hb[32][SPW_LD];
  __shared__ float lg[32][MAX_N];
  const int tid  = threadIdx.x;
  const int row0 = blockIdx.x * 32;

  for (int i = tid; i < 4096; i += 256) {
    int r = i >> 7, cq = i & 127;
    ((float4*)&zt[r][0])[cq] = ((const float4*)(z + (size_t)(row0 + r) * HID))[cq];
  }
  __syncthreads();

  const int wave = tid >> 5, lane = tid & 31, half = lane >> 4, l16 = lane & 15;

  // GEMM1: [32x512] @ [512x272(packed,padded)]; dual-M accumulators share B.
  for (int tile = wave; tile < 17; tile += 8) {
    int col = tile * 16 + l16;
    int voff = 2 * col + 2 * SPW_LD * half;
    v8f acc0 = {}, acc1 = {};
    gemm_dual<SPW_LD, HID>(W1p, voff, &zt[l16][0], &zt[16 + l16][0], 2 * half, acc0, acc1);
    if (col < SP_MID) {
      float bias = b1[col];
      #pragma unroll
      for (int v = 0; v < 8; v++) {
        hb[v + 8 * half][col]      = acc0[v] + bias;
        hb[16 + v + 8 * half][col] = acc1[v] + bias;
      }
    }
  }
  __syncthreads();

  // LayerNorm (biased var, eps=1e-5) + ReLU; 16 threads per row, 2 row passes.
  for (int rr = tid >> 4; rr < 32; rr += 16) {
    int c16 = tid & 15;
    float s = 0.f, ss = 0.f;
    for (int j = 0; j < 17; j++) {
      int c = c16 + 16 * j;
      if (c < SP_MID) { float v = hb[rr][c]; s += v; ss += v * v; }
    }
    for (int m = 8; m >= 1; m >>= 1) { s += __shfl_xor(s, m, 16); ss += __shfl_xor(ss, m, 16); }
    float mean = s * (1.0f / SP_MID);
    float var  = ss * (1.0f / SP_MID) - mean * mean;
    float inv  = rsqrtf(var + 1e-5f);
    for (int j = 0; j < 17; j++) {
      int c = c16 + 16 * j;
      if (c < SP_MID) {
        float v = (hb[rr][c] - mean) * inv * g[c] + be[c];
        hb[rr][c] = v > 0.f ? v : 0.f;
      }
    }
  }
  __syncthreads();

  // GEMM2 (tiny, VALU): logits[32x17]
  for (int idx = tid; idx < 32 * MAX_N; idx += 256) {
    int r = idx / MAX_N, c = idx % MAX_N;
    float acc = b2[c];
    for (int j = 0; j < SP_MID; j++) acc += hb[r][j] * W2[j * MAX_N + c];
    lg[r][c] = acc;
  }
  __syncthreads();

  if (tid < 32) {
    float best = lg[tid][0]; int bi = 0;
    for (int c = 1; c < MAX_N; c++) { float v = lg[tid][c]; if (v > best) { best = v; bi = c; } }
    n_out[row0 + tid] = bi;
  }
}

// ---------------------------------------------------------------------------
// Phase 2: exclusive scan of n -> starts, total T. Single block, 1024 thr x 32.
// ---------------------------------------------------------------------------
__global__ __launch_bounds__(1024) void scan_kernel(
    const int* __restrict__ n, int* __restrict__ starts, int* __restrict__ total)
{
  __shared__ int sums[1024];
  int tid = threadIdx.x;
  int base = tid * 32;
  int s = 0;
  for (int j = 0; j < 32; j++) s += n[base + j];
  sums[tid] = s;
  __syncthreads();
  for (int off = 1; off < 1024; off <<= 1) {
    int v = (tid >= off) ? sums[tid - off] : 0;
    __syncthreads();
    sums[tid] += v;
    __syncthreads();
  }
  int run = (tid == 0) ? 0 : sums[tid - 1];
  for (int j = 0; j < 32; j++) { int c = n[base + j]; starts[base + j] = run; run += c; }
  if (tid == 1023) *total = sums[1023];
}

// ---------------------------------------------------------------------------
// Phase 3: key table. pos is one-hot -> only 17 distinct keys:
//   keys[k] = relu(LN(key_W1[k] + b1)) @ key_W2 + b2.   (tiny; VALU)
// ---------------------------------------------------------------------------
__global__ __launch_bounds__(256) void keys_kernel(
    const float* __restrict__ W1, const float* __restrict__ b1,
    const float* __restrict__ g, const float* __restrict__ be,
    const float* __restrict__ W2, const float* __restrict__ b2,
    float* __restrict__ keys)
{
  __shared__ float h[KEY_MID];
  __shared__ float red[256];
  const int k = blockIdx.x;
  const int tid = threadIdx.x;
  float s = 0.f, ss = 0.f;
  for (int c = tid; c < KEY_MID; c += 256) {
    float v = W1[k * KEY_MID + c] + b1[c];
    h[c] = v; s += v; ss += v * v;
  }
  red[tid] = s; __syncthreads();
  for (int off = 128; off >= 1; off >>= 1) { if (tid < off) red[tid] += red[tid + off]; __syncthreads(); }
  float sum = red[0]; __syncthreads();
  red[tid] = ss; __syncthreads();
  for (int off = 128; off >= 1; off >>= 1) { if (tid < off) red[tid] += red[tid + off]; __syncthreads(); }
  float sumsq = red[0]; __syncthreads();
  float mean = sum * (1.0f / KEY_MID);
  float var  = sumsq * (1.0f / KEY_MID) - mean * mean;
  float inv  = rsqrtf(var + 1e-5f);
  for (int c = tid; c < KEY_MID; c += 256) {
    float v = (h[c] - mean) * inv * g[c] + be[c];
    h[c] = v > 0.f ? v : 0.f;
  }
  __syncthreads();
  for (int c = tid; c < HID; c += 256) {
    float acc = b2[c];
    for (int j = 0; j < KEY_MID; j++) acc += h[j] * W2[j * HID + c];
    keys[k * HID + c] = acc;
  }
}

// ---------------------------------------------------------------------------
// Phase 4: materialize (batch, k) per token.
// ---------------------------------------------------------------------------
__global__ __launch_bounds__(256) void token_kernel(
    const int* __restrict__ n, const int* __restrict__ starts,
    int* __restrict__ tokb, int* __restrict__ tokk)
{
  int i = blockIdx.x * 256 + threadIdx.x;
  int st = starts[i]; int c = n[i];
  for (int j = 0; j < c; j++) { tokb[st + j] = i; tokk[st + j] = j; }
}

// ---------------------------------------------------------------------------
// Phase 5: decoder. 32 tokens per block (two 16-row M tiles), 8 waves.
//   zp = z[batch]*keys[k];  mid = relu(zp@W1+b1);  x = mid@W2+b2
// ---------------------------------------------------------------------------
#define ZP_LD  516
#define MID_LD 388

__global__ __launch_bounds__(256) void decoder_kernel(
    const float* __restrict__ z, const float* __restrict__ keys,
    const int* __restrict__ tokb, const int* __restrict__ tokk,
    const float* __restrict__ W1p, const float* __restrict__ b1,
    const float* __restrict__ W2p, const float* __restrict__ b2,
    const int* __restrict__ totalp, int Thost, float* __restrict__ out)
{
  __shared__ float zp[32][ZP_LD];
  __shared__ float mid[32][MID_LD];
  const int T  = min(*totalp, Thost);   // Thost-clamped: never writes past d_out
  const int t0 = blockIdx.x * 32;
  if (t0 >= T) return;
  const int tid = threadIdx.x;

  // stage A: zp[r][:] = z[batch[t]] * keys[k[t]]  (float4, zero-fill tail rows)
  for (int i = tid; i < 4096; i += 256) {
    int r = i >> 7, cq = i & 127;
    int t = t0 + r;
    float4 p = {0.f, 0.f, 0.f, 0.f};
    if (t < T) {
      int b = tokb[t], kp = tokk[t];
      float4 zv = ((const float4*)(z    + (size_t)b  * HID))[cq];
      float4 kv = ((const float4*)(keys + (size_t)kp * HID))[cq];
      p.x = zv.x * kv.x; p.y = zv.y * kv.y; p.z = zv.z * kv.z; p.w = zv.w * kv.w;
    }
    ((float4*)&zp[r][0])[cq] = p;
  }
  __syncthreads();

  const int wave = tid >> 5, lane = tid & 31, half = lane >> 4, l16 = lane & 15;

  // GEMM1: [32x512] @ [512x384] + bias, ReLU -> mid   (24 N-tiles / 8 waves)
  for (int tile = wave; tile < DEC_MID / 16; tile += 8) {
    int col = tile * 16 + l16;
    int voff = 2 * col + 2 * DEC_MID * half;
    v8f acc0 = {}, acc1 = {};
    gemm_dual<DEC_MID, HID>(W1p, voff, &zp[l16][0], &zp[16 + l16][0], 2 * half, acc0, acc1);
    float bias = b1[col];
    #pragma unroll
    for (int v = 0; v < 8; v++) {
      float x0 = acc0[v] + bias;
      float x1 = acc1[v] + bias;
      mid[v + 8 * half][col]      = x0 > 0.f ? x0 : 0.f;
      mid[16 + v + 8 * half][col] = x1 > 0.f ? x1 : 0.f;
    }
  }
  __syncthreads();

  // GEMM2: [32x384] @ [384x256] + bias -> out   (16 N-tiles / 8 waves)
  for (int tile = wave; tile < DIM / 16; tile += 8) {
    int col = tile * 16 + l16;
    int voff = 2 * col + 2 * DIM * half;
    v8f acc0 = {}, acc1 = {};
    gemm_dual<DIM, DEC_MID>(W2p, voff, &mid[l16][0], &mid[16 + l16][0], 2 * half, acc0, acc1);
    float bias = b2[col];
    #pragma unroll
    for (int v = 0; v < 8; v++) {
      int t0v = t0 + v + 8 * half;
      int t1v = t0 + 16 + v + 8 * half;
      if (t0v < T) out[(size_t)t0v * DIM + col] = acc0[v] + bias;
      if (t1v < T) out[(size_t)t1v * DIM + col] = acc1[v] + bias;
    }
  }

  // second output: batch ids as float, at the harness's layout offset
  if (tid < 32) {
    int t = t0 + tid;
    if (t < T) out[(size_t)Thost * DIM + t] = (float)tokb[t];
  }
}

// ---------------------------------------------------------------------------
extern "C" void kernel_launch(void* const* d_in, const int* in_sizes, int n_in,
                              void* d_out, int out_size, void* d_ws, size_t ws_size,
                              hipStream_t stream) {
  (void)in_sizes; (void)n_in; (void)ws_size;
  const float* z       = (const float*)d_in[0];
  const float* key_W1  = (const float*)d_in[1];
  const float* key_b1  = (const float*)d_in[2];
  const float* key_g1  = (const float*)d_in[3];
  const float* key_be1 = (const float*)d_in[4];
  const float* key_W2  = (const float*)d_in[5];
  const float* key_b2  = (const float*)d_in[6];
  const float* dec_W1  = (const float*)d_in[7];
  const float* dec_b1  = (const float*)d_in[8];
  const float* dec_W2  = (const float*)d_in[9];
  const float* dec_b2  = (const float*)d_in[10];
  const float* sp_W1   = (const float*)d_in[11];
  const float* sp_b1   = (const float*)d_in[12];
  const float* sp_g1   = (const float*)d_in[13];
  const float* sp_be1  = (const float*)d_in[14];
  const float* sp_W2   = (const float*)d_in[15];
  const float* sp_b2   = (const float*)d_in[16];
  float* out = (float*)d_out;

  char* ws = (char*)d_ws;
  size_t off = 0;
  int*   d_n      = (int*)(ws + off);   off += (size_t)B_ROWS * 4;
  int*   d_starts = (int*)(ws + off);   off += (size_t)B_ROWS * 4;
  int*   d_total  = (int*)(ws + off);   off += 16;
  float* d_keys   = (float*)(ws + off); off += (size_t)MAX_N * HID * 4;
  float* d_spW1p  = (float*)(ws + off); off += (size_t)HID * SPW_LD * 4;
  float* d_decW1p = (float*)(ws + off); off += (size_t)HID * DEC_MID * 4;
  float* d_decW2p = (float*)(ws + off); off += (size_t)DEC_MID * DIM * 4;
  int*   d_tokb   = (int*)(ws + off);   off += (size_t)B_ROWS * 16 * 4;
  int*   d_tokk   = (int*)(ws + off);   off += (size_t)B_ROWS * 16 * 4;

  const int Thost = out_size / (DIM + 1);   // x is [T,256] + batch [T] -> T*257 floats

  pack_w_kernel<<<(HID * SPW_LD + 255) / 256, 256, 0, stream>>>(sp_W1, d_spW1p, HID, SP_MID, SPW_LD);
  pack_w_kernel<<<(HID * DEC_MID + 255) / 256, 256, 0, stream>>>(dec_W1, d_decW1p, HID, DEC_MID, DEC_MID);
  pack_w_kernel<<<(DEC_MID * DIM + 255) / 256, 256, 0, stream>>>(dec_W2, d_decW2p, DEC_MID, DIM, DIM);
  size_pred_kernel<<<B_ROWS / 32, 256, 0, stream>>>(z, d_spW1p, sp_b1, sp_g1, sp_be1, sp_W2, sp_b2, d_n);
  scan_kernel<<<1, 1024, 0, stream>>>(d_n, d_starts, d_total);
  keys_kernel<<<MAX_N, 256, 0, stream>>>(key_W1, key_b1, key_g1, key_be1, key_W2, key_b2, d_keys);
  token_kernel<<<B_ROWS / 256, 256, 0, stream>>>(d_n, d_starts, d_tokb, d_tokk);
  decoder_kernel<<<B_ROWS * 16 / 32, 256, 0, stream>>>(z, d_keys, d_tokb, d_tokk,
                                                       d_decW1p, dec_b1, d_decW2p, dec_b2,
                                                       d_total, Thost, out);
}